// SE3TransformerWrapper_8100308320264
// MI455X (gfx1250) — compile-verified
//
#include <hip/hip_runtime.h>
#include <math.h>

// ---------------------------------------------------------------------------
// SE(3)-Transformer forward for MI455X (gfx1250, wave32).
//  - Edge work (gather/scatter bound): fused radial-MLP + value + atomic
//    scatter; weights staged TRANSPOSED in LDS so every dot product reads
//    consecutive floats (ds_load_b128-friendly); row gathers use b128 loads.
//  - Node GEMMs (K=32, M=20000=1250*16): V_WMMA_F32_16X16X4_F32, one 16x16
//    output tile per wave, 8 WMMA k-steps; residual templated (branch-free).
//  - Softmax over destination: atomicMax (int-trick) + atomicAdd segments.
// ---------------------------------------------------------------------------

typedef __attribute__((ext_vector_type(2))) float v2f;
typedef __attribute__((ext_vector_type(8))) float v8f;

#define NNODES 20000
#define NEDGES 400000

static __device__ __forceinline__ void atomic_max_f32(float* addr, float val) {
  // works for mixed signs when initialized to -inf
  if (val >= 0.0f) atomicMax(reinterpret_cast<int*>(addr), __float_as_int(val));
  else             atomicMin(reinterpret_cast<unsigned int*>(addr), __float_as_uint(val));
}

static __device__ __forceinline__ void load_row32(float (&dst)[32],
                                                  const float* __restrict__ p) {
  const float4* p4 = reinterpret_cast<const float4*>(p);
#pragma unroll
  for (int w = 0; w < 8; ++w) {
    const float4 t = p4[w];
    dst[4 * w + 0] = t.x; dst[4 * w + 1] = t.y;
    dst[4 * w + 2] = t.z; dst[4 * w + 3] = t.w;
  }
}

static __device__ __forceinline__ float dot32(const float (&a)[32],
                                              const float* __restrict__ b) {
  float acc = 0.f;
#pragma unroll
  for (int i = 0; i < 32; ++i) acc += a[i] * b[i];
  return acc;
}

// --------------------------- utility fill ----------------------------------
__global__ void fill_kernel(float* __restrict__ p, float v, long long n) {
  long long i  = (long long)blockIdx.x * blockDim.x + threadIdx.x;
  long long st = (long long)gridDim.x * blockDim.x;
  for (; i < n; i += st) p[i] = v;
}

// --------------------- WMMA GEMM: C = A(M,32) @ B(32,N) (+R) ---------------
// Row-major everywhere. M % 16 == 0, N % 16 == 0. One 16x16 tile per wave.
template <bool HASR>
__global__ __launch_bounds__(256) void gemm_nk32_wmma(
    const float* __restrict__ A, const float* __restrict__ B,
    const float* __restrict__ R, float* __restrict__ C, int M, int N) {
  const int lane = threadIdx.x & 31;
  const int wave = threadIdx.x >> 5;
  const int tilesN = N >> 4;
  const int tilesM = M >> 4;
  int gid = blockIdx.x * 8 + wave;
  if (gid >= tilesM * tilesN) return;
  const int mt = gid / tilesN, nt = gid % tilesN;
  const int row = mt * 16 + (lane & 15);   // A row for this lane
  const int col = nt * 16 + (lane & 15);   // B/D column for this lane
  const int kh  = (lane >> 4) * 2;         // lanes 16..31 hold K+2,K+3
  union { v8f v; float f[8]; } acc;
#pragma unroll
  for (int i = 0; i < 8; ++i) acc.f[i] = 0.0f;
  const float* arow = A + (long long)row * 32;
#pragma unroll
  for (int s = 0; s < 8; ++s) {
    const int k0 = 4 * s + kh;
    v2f a, b;
    a.x = arow[k0];
    a.y = arow[k0 + 1];
    b.x = B[(long long)k0 * N + col];
    b.y = B[(long long)(k0 + 1) * N + col];
    acc.v = __builtin_amdgcn_wmma_f32_16x16x4_f32(
        false, a, false, b, (short)0, acc.v, false, false);
  }
  const int rbase = mt * 16 + (lane >> 4) * 8;
#pragma unroll
  for (int i = 0; i < 8; ++i) {
    const long long idx = (long long)(rbase + i) * N + col;
    float v = acc.f[i];
    if (HASR) v += R[idx];
    C[idx] = v;
  }
}

// ------------------- readout GEMM: sigmoid(h0 @ Bc) -> out[t][n][k] --------
// A: (20000,32); Bc: (32,240) (cols 225..239 are zero pad); out: 15x20000x15
__global__ __launch_bounds__(256) void cs_gemm_wmma(
    const float* __restrict__ A, const float* __restrict__ B,
    float* __restrict__ out) {
  const int lane = threadIdx.x & 31;
  const int wave = threadIdx.x >> 5;
  int gid = blockIdx.x * 8 + wave;
  if (gid >= 1250 * 15) return;
  const int mt = gid / 15, nt = gid % 15;
  const int row = mt * 16 + (lane & 15);
  const int col = nt * 16 + (lane & 15);
  const int kh  = (lane >> 4) * 2;
  union { v8f v; float f[8]; } acc;
#pragma unroll
  for (int i = 0; i < 8; ++i) acc.f[i] = 0.0f;
  const float* arow = A + (long long)row * 32;
#pragma unroll
  for (int s = 0; s < 8; ++s) {
    const int k0 = 4 * s + kh;
    v2f a, b;
    a.x = arow[k0];
    a.y = arow[k0 + 1];
    b.x = B[k0 * 240 + col];
    b.y = B[(k0 + 1) * 240 + col];
    acc.v = __builtin_amdgcn_wmma_f32_16x16x4_f32(
        false, a, false, b, (short)0, acc.v, false, false);
  }
  if (col < 225) {
    const int t = col / 15, k = col % 15;
    const int rbase = mt * 16 + (lane >> 4) * 8;
#pragma unroll
    for (int i = 0; i < 8; ++i) {
      const int rr = rbase + i;
      const float v = acc.f[i];
      out[((long long)t * NNODES + rr) * 15 + k] = 1.0f / (1.0f + __expf(-v));
    }
  }
}

// -------- repack Wc (15,15,32) -> Bc (32,240), zero pad cols 225..239 ------
__global__ void repack_wc_kernel(const float* __restrict__ Wc,
                                 float* __restrict__ Bc) {
  int idx = blockIdx.x * 256 + threadIdx.x;  // d*240 + j
  if (idx >= 32 * 240) return;
  const int d = idx / 240, j = idx % 240;
  Bc[idx] = (j < 225) ? Wc[j * 32 + d] : 0.0f;
}

// ------------------------- edge pass 1: logits -----------------------------
__global__ __launch_bounds__(256) void edge_logits_kernel(
    const float* __restrict__ f0, const float* __restrict__ ef,
    const int* __restrict__ src, const int* __restrict__ dst,
    const float* __restrict__ q, const float* __restrict__ Wk,
    float* __restrict__ lg, float* __restrict__ mx, int E) {
  __shared__ float sWkT[32 * 64];  // [j][i], i over 64 inputs
  for (int idx = threadIdx.x; idx < 64 * 32; idx += 256) {
    const int i = idx >> 5, j = idx & 31;
    sWkT[j * 64 + i] = Wk[idx];
  }
  __syncthreads();
  const int e = blockIdx.x * 256 + threadIdx.x;
  if (e >= E) return;
  const int s = src[e], d = dst[e];
  float fs[32], ev[32], qd[32];
  load_row32(fs, f0 + (long long)s * 32);
  load_row32(ev, ef + (long long)e * 32);
  load_row32(qd, q + (long long)d * 32);
  float lh[4] = {0.f, 0.f, 0.f, 0.f};
  for (int j = 0; j < 32; ++j) {
    const float* wr = &sWkT[j * 64];
    float kj = 0.f;
#pragma unroll
    for (int i = 0; i < 32; ++i) kj += fs[i] * wr[i] + ev[i] * wr[32 + i];
    lh[j >> 3] += qd[j] * kj;
  }
#pragma unroll
  for (int h = 0; h < 4; ++h) {
    const float v = lh[h] * 0.3535533906f;  // 1/sqrt(8)
    lg[(long long)e * 4 + h] = v;
    atomic_max_f32(&mx[d * 4 + h], v);
  }
}

// ------------------------- edge pass 2: exp / denom ------------------------
__global__ void edge_exp_kernel(float* __restrict__ lg,
                                const float* __restrict__ mx,
                                const int* __restrict__ dst,
                                float* __restrict__ den, int E) {
  const int e = blockIdx.x * 256 + threadIdx.x;
  if (e >= E) return;
  const int d = dst[e];
  float4 l = *reinterpret_cast<const float4*>(lg + (long long)e * 4);
  const float4 m4 = *reinterpret_cast<const float4*>(mx + (long long)d * 4);
  l.x = __expf(l.x - m4.x);
  l.y = __expf(l.y - m4.y);
  l.z = __expf(l.z - m4.z);
  l.w = __expf(l.w - m4.w);
  *reinterpret_cast<float4*>(lg + (long long)e * 4) = l;
  atomicAdd(&den[d * 4 + 0], l.x);
  atomicAdd(&den[d * 4 + 1], l.y);
  atomicAdd(&den[d * 4 + 2], l.z);
  atomicAdd(&den[d * 4 + 3], l.w);
}

// ---------------- edge pass 3: fused radial MLP + value + scatter ----------
template <int LAYER>
__global__ __launch_bounds__(256) void edge_value_kernel(
    const float* __restrict__ pos, const float* __restrict__ ef,
    const int* __restrict__ src, const int* __restrict__ dst,
    const float* __restrict__ f0,
    const float* __restrict__ feats1, const float* __restrict__ feats2,
    const float* __restrict__ ex, const float* __restrict__ den,
    const float* __restrict__ Wr1, const float* __restrict__ br1,
    const float* __restrict__ Wr2, const float* __restrict__ br2,
    const float* __restrict__ Wv00, const float* __restrict__ Wv01,
    const float* __restrict__ Wv02, const float* __restrict__ Wv11,
    const float* __restrict__ Wv22,
    float* __restrict__ agg0, float* __restrict__ agg1,
    float* __restrict__ agg2, int E) {
  constexpr int RAD = LAYER ? 160 : 96;
  // transposed weight tiles: inner dots read consecutive floats
  __shared__ float sWr1T[32 * 36];   // [j][i], stride 36 keeps 16B alignment
  __shared__ float sbr1[32];
  __shared__ float sWr2T[160 * 32];  // [col][i]
  __shared__ float sbr2[160];
  __shared__ float sWv0T[3 * 1024];  // [lo][o][i]
  __shared__ float sWv11[1024];      // row-major [i][o] (o-contiguous use)
  __shared__ float sWv22[1024];
  for (int idx = threadIdx.x; idx < 33 * 32; idx += 256) {
    const int i = idx >> 5, j = idx & 31;
    sWr1T[j * 36 + i] = Wr1[idx];
  }
  for (int i = threadIdx.x; i < 32; i += 256) sbr1[i] = br1[i];
  for (int idx = threadIdx.x; idx < 32 * RAD; idx += 256) {
    const int i = idx / RAD, c = idx % RAD;
    sWr2T[c * 32 + i] = Wr2[idx];
  }
  for (int i = threadIdx.x; i < RAD; i += 256) sbr2[i] = br2[i];
  for (int idx = threadIdx.x; idx < 1024; idx += 256) {
    const int i = idx >> 5, o = idx & 31;
    sWv0T[o * 32 + i]        = Wv00[idx];
    sWv0T[1024 + o * 32 + i] = Wv01[idx];
    sWv0T[2048 + o * 32 + i] = Wv02[idx];
    if (LAYER) { sWv11[idx] = Wv11[idx]; sWv22[idx] = Wv22[idx]; }
  }
  __syncthreads();
  const int e = blockIdx.x * 256 + threadIdx.x;
  if (e >= E) return;
  const int sN = src[e], dN = dst[e];
  const float px = pos[dN * 3 + 0] - pos[sN * 3 + 0];
  const float py = pos[dN * 3 + 1] - pos[sN * 3 + 1];
  const float pz = pos[dN * 3 + 2] - pos[sN * 3 + 2];
  const float r = sqrtf(px * px + py * py + pz * pz + 1e-12f);
  const float inv = 1.0f / r;
  const float x = px * inv, y = py * inv, z = pz * inv;
  const float SQ3 = 1.7320508075688772f;
  const float Y1[3] = {x, y, z};
  const float Y2[5] = {SQ3 * x * y, SQ3 * y * z, 0.5f * (3.f * z * z - 1.f),
                       SQ3 * x * z, 0.5f * SQ3 * (x * x - y * y)};
  float a4[4];
#pragma unroll
  for (int h = 0; h < 4; ++h)
    a4[h] = ex[(long long)e * 4 + h] / (den[dN * 4 + h] + 1e-9f);
  float ev[32], fs[32], hh[32];
  load_row32(ev, ef + (long long)e * 32);
  load_row32(fs, f0 + (long long)sN * 32);
  // radial hidden: h = relu([ef, r] @ Wr1 + br1)
  for (int j = 0; j < 32; ++j) {
    const float* wr = &sWr1T[j * 36];
    const float acc = sbr1[j] + r * wr[32] + dot32(ev, wr);
    hh[j] = acc > 0.f ? acc : 0.f;
  }
  // ----- lo = 0 -----
  for (int o = 0; o < 32; ++o) {
    const float g = sbr2[o] + dot32(hh, &sWr2T[o * 32]);
    const float s = dot32(fs, &sWv0T[o * 32]);
    atomicAdd(&agg0[dN * 32 + o], s * g * a4[o >> 3]);
  }
  // ----- lo = 1 -----
  {
    const int GC = 32, G2C = 64;
    float s1[32], g2[32];
    for (int o = 0; o < 32; ++o) {
      const float g = sbr2[GC + o] + dot32(hh, &sWr2T[(GC + o) * 32]);
      s1[o] = dot32(fs, &sWv0T[1024 + o * 32]) * g;
      if (LAYER) g2[o] = sbr2[G2C + o] + dot32(hh, &sWr2T[(G2C + o) * 32]);
    }
    for (int m = 0; m < 3; ++m) {
      float u[32];
      if (LAYER) {
#pragma unroll
        for (int o = 0; o < 32; ++o) u[o] = 0.f;
        float flr[32];
        load_row32(flr, feats1 + ((long long)m * NNODES + sN) * 32);
        for (int i = 0; i < 32; ++i) {
          const float fl = flr[i];
#pragma unroll
          for (int o = 0; o < 32; ++o) u[o] += fl * sWv11[i * 32 + o];
        }
      }
      for (int o = 0; o < 32; ++o) {
        float v = s1[o] * Y1[m];
        if (LAYER) v += u[o] * g2[o];
        atomicAdd(&agg1[((long long)m * NNODES + dN) * 32 + o], v * a4[o >> 3]);
      }
    }
  }
  // ----- lo = 2 -----
  {
    const int GC = LAYER ? 96 : 64, G2C = 128;
    float s1[32], g2[32];
    for (int o = 0; o < 32; ++o) {
      const float g = sbr2[GC + o] + dot32(hh, &sWr2T[(GC + o) * 32]);
      s1[o] = dot32(fs, &sWv0T[2048 + o * 32]) * g;
      if (LAYER) g2[o] = sbr2[G2C + o] + dot32(hh, &sWr2T[(G2C + o) * 32]);
    }
    for (int m = 0; m < 5; ++m) {
      float u[32];
      if (LAYER) {
#pragma unroll
        for (int o = 0; o < 32; ++o) u[o] = 0.f;
        float flr[32];
        load_row32(flr, feats2 + ((long long)m * NNODES + sN) * 32);
        for (int i = 0; i < 32; ++i) {
          const float fl = flr[i];
#pragma unroll
          for (int o = 0; o < 32; ++o) u[o] += fl * sWv22[i * 32 + o];
        }
      }
      for (int o = 0; o < 32; ++o) {
        float v = s1[o] * Y2[m];
        if (LAYER) v += u[o] * g2[o];
        atomicAdd(&agg2[((long long)m * NNODES + dN) * 32 + o], v * a4[o >> 3]);
      }
    }
  }
}

// ---------------------------------------------------------------------------
struct LayerW {
  const float *Wk, *Wq, *Wr1, *Wr2, *Ws0, *Ws1, *Ws2;
  const float *Wv00, *Wv01, *Wv02, *Wv11, *Wv22, *br1, *br2;
};

extern "C" void kernel_launch(void* const* d_in, const int* in_sizes, int n_in,
                              void* d_out, int out_size, void* d_ws,
                              size_t ws_size, hipStream_t stream) {
  const float* node_feats = (const float*)d_in[0];
  const float* pos        = (const float*)d_in[1];
  const float* edge_feats = (const float*)d_in[2];
  const int*   src        = (const int*)d_in[3];
  const int*   dst        = (const int*)d_in[4];

  // Params: either one flat blob (n_in==6) or one pointer per leaf, in
  // recursive dict-insertion order of the reference's param construction.
  LayerW L0, L1;
  const float *Wout0, *Wc;
  if (n_in >= 33) {
    int i = 5;
    // layer0 insertion order: Wv00,Ws0,Wv01,Ws1,Wv02,Ws2,Wr1,br1,Wr2,br2,Wq,Wk
    L0.Wv00 = (const float*)d_in[i++]; L0.Ws0 = (const float*)d_in[i++];
    L0.Wv01 = (const float*)d_in[i++]; L0.Ws1 = (const float*)d_in[i++];
    L0.Wv02 = (const float*)d_in[i++]; L0.Ws2 = (const float*)d_in[i++];
    L0.Wr1  = (const float*)d_in[i++]; L0.br1 = (const float*)d_in[i++];
    L0.Wr2  = (const float*)d_in[i++]; L0.br2 = (const float*)d_in[i++];
    L0.Wq   = (const float*)d_in[i++]; L0.Wk  = (const float*)d_in[i++];
    L0.Wv11 = nullptr; L0.Wv22 = nullptr;
    // layer1: Wv00,Ws0,Wv01,Wv11,Ws1,Wv02,Wv22,Ws2,Wr1,br1,Wr2,br2,Wq,Wk
    L1.Wv00 = (const float*)d_in[i++]; L1.Ws0  = (const float*)d_in[i++];
    L1.Wv01 = (const float*)d_in[i++]; L1.Wv11 = (const float*)d_in[i++];
    L1.Ws1  = (const float*)d_in[i++]; L1.Wv02 = (const float*)d_in[i++];
    L1.Wv22 = (const float*)d_in[i++]; L1.Ws2  = (const float*)d_in[i++];
    L1.Wr1  = (const float*)d_in[i++]; L1.br1  = (const float*)d_in[i++];
    L1.Wr2  = (const float*)d_in[i++]; L1.br2  = (const float*)d_in[i++];
    L1.Wq   = (const float*)d_in[i++]; L1.Wk   = (const float*)d_in[i++];
    Wout0 = (const float*)d_in[i++];
    Wc    = (const float*)d_in[i++];
  } else {
    const float* p = (const float*)d_in[5];
    size_t o = 0;
    auto take = [&](size_t n) { const float* r = p + o; o += n; return r; };
    L0.Wv00 = take(1024); L0.Ws0 = take(1024);
    L0.Wv01 = take(1024); L0.Ws1 = take(1024);
    L0.Wv02 = take(1024); L0.Ws2 = take(1024);
    L0.Wr1  = take(33 * 32); L0.br1 = take(32);
    L0.Wr2  = take(32 * 96); L0.br2 = take(96);
    L0.Wq   = take(1024); L0.Wk = take(2048);
    L0.Wv11 = nullptr; L0.Wv22 = nullptr;
    L1.Wv00 = take(1024); L1.Ws0 = take(1024);
    L1.Wv01 = take(1024); L1.Wv11 = take(1024);
    L1.Ws1  = take(1024); L1.Wv02 = take(1024);
    L1.Wv22 = take(1024); L1.Ws2 = take(1024);
    L1.Wr1  = take(33 * 32); L1.br1 = take(32);
    L1.Wr2  = take(32 * 160); L1.br2 = take(160);
    L1.Wq   = take(1024); L1.Wk = take(2048);
    Wout0 = take(1024);
    Wc    = take(15 * 15 * 32);
  }

  // workspace layout (floats)
  float* ws = (float*)d_ws;
  float* q    = ws;                               // 640000
  float* lg   = ws + 640000;                      // 1600000 (logits -> exp)
  float* mx   = ws + 2240000;                     // 80000
  float* den  = ws + 2320000;                     // 80000
  float* agg0 = ws + 2400000;                     // 640000
  float* agg1 = ws + 3040000;                     // 1920000
  float* agg2 = ws + 4960000;                     // 3200000
  float* fa0  = ws + 8160000;                     // 640000 (layer0 out, lo=0)
  float* fa1  = ws + 8800000;                     // 1920000
  float* fa2  = ws + 10720000;                    // 3200000
  float* fb0  = ws + 13920000;                    // 640000 (layer1 out, lo=0)
  float* h0   = ws + 14560000;                    // 640000
  float* Bc   = ws + 15200000;                    // 7680

  const int eb = (NEDGES + 255) / 256;
  auto gemm = [&](const float* A, const float* B, const float* R, float* C,
                  int M, int N) {
    const int tiles = (M / 16) * (N / 16);
    const int blocks = (tiles + 7) / 8;
    if (R)
      gemm_nk32_wmma<true><<<blocks, 256, 0, stream>>>(A, B, R, C, M, N);
    else
      gemm_nk32_wmma<false><<<blocks, 256, 0, stream>>>(A, B, R, C, M, N);
  };
  auto fill = [&](float* p, float v, long long n) {
    int blocks = (int)((n + 255) / 256);
    if (blocks > 4096) blocks = 4096;
    fill_kernel<<<blocks, 256, 0, stream>>>(p, v, n);
  };

  // ============================== layer 0 ==============================
  gemm(node_feats, L0.Wq, nullptr, q, NNODES, 32);
  fill(mx, -INFINITY, NNODES * 4);
  fill(den, 0.0f, NNODES * 4);
  fill(agg0, 0.0f, (long long)NNODES * 32 * 9);  // agg0..agg2 contiguous
  edge_logits_kernel<<<eb, 256, 0, stream>>>(node_feats, edge_feats, src, dst,
                                             q, L0.Wk, lg, mx, NEDGES);
  edge_exp_kernel<<<eb, 256, 0, stream>>>(lg, mx, dst, den, NEDGES);
  edge_value_kernel<0><<<eb, 256, 0, stream>>>(
      pos, edge_feats, src, dst, node_feats, nullptr, nullptr, lg, den,
      L0.Wr1, L0.br1, L0.Wr2, L0.br2, L0.Wv00, L0.Wv01, L0.Wv02, nullptr,
      nullptr, agg0, agg1, agg2, NEDGES);
  gemm(agg0, L0.Ws0, node_feats, fa0, NNODES, 32);  // residual (lo=0 only)
  for (int m = 0; m < 3; ++m)
    gemm(agg1 + (long long)m * NNODES * 32, L0.Ws1, nullptr,
         fa1 + (long long)m * NNODES * 32, NNODES, 32);
  for (int m = 0; m < 5; ++m)
    gemm(agg2 + (long long)m * NNODES * 32, L0.Ws2, nullptr,
         fa2 + (long long)m * NNODES * 32, NNODES, 32);

  // ============================== layer 1 ==============================
  gemm(fa0, L1.Wq, nullptr, q, NNODES, 32);
  fill(mx, -INFINITY, NNODES * 4);
  fill(den, 0.0f, NNODES * 4);
  fill(agg0, 0.0f, (long long)NNODES * 32 * 9);
  edge_logits_kernel<<<eb, 256, 0, stream>>>(fa0, edge_feats, src, dst, q,
                                             L1.Wk, lg, mx, NEDGES);
  edge_exp_kernel<<<eb, 256, 0, stream>>>(lg, mx, dst, den, NEDGES);
  edge_value_kernel<1><<<eb, 256, 0, stream>>>(
      pos, edge_feats, src, dst, fa0, fa1, fa2, lg, den, L1.Wr1, L1.br1,
      L1.Wr2, L1.br2, L1.Wv00, L1.Wv01, L1.Wv02, L1.Wv11, L1.Wv22, agg0, agg1,
      agg2, NEDGES);
  // readout only needs lo=0 of layer1 -> skip Ws1/Ws2 stages
  gemm(agg0, L1.Ws0, fa0, fb0, NNODES, 32);  // + residual fa0

  // ============================== readout ==============================
  gemm(fb0, Wout0, nullptr, h0, NNODES, 32);
  repack_wc_kernel<<<(32 * 240 + 255) / 256, 256, 0, stream>>>(Wc, Bc);
  cs_gemm_wmma<<<(1250 * 15 + 7) / 8, 256, 0, stream>>>(h0, Bc, (float*)d_out);
}